// my_encoding_cubic_79783312490827
// MI455X (gfx1250) — compile-verified
//
#include <hip/hip_runtime.h>
#include <stdint.h>

// Multiresolution 2-D hash encoding, bicubic-Hermite == Catmull-Rom form.
// out[n, l*2+f] = sum_{i,j} u_i(tx) v_j(ty) * HT[hash(ix+i, iy+j), f, l]
// Gather-bound: 134M random dword loads, working set ~93MB (L2-resident).
// Outputs staged in LDS and drained with gfx1250 async store-from-LDS (coalesced b128).

#define TMASK   0x3FFFFFu          // T = 2^22
#define HPRIME  2654435761u
#define LVLS    16
#define BLK     256

__constant__ float c_NL[LVLS] = {
    16.f, 20.f, 25.f, 32.f, 40.f, 50.f, 64.f, 80.f,
    101.f, 128.f, 161.f, 203.f, 256.f, 322.f, 406.f, 512.f
};

typedef int v4i __attribute__((__vector_size__(4 * sizeof(int))));
typedef __attribute__((address_space(3))) v4i   lds_v4i_t;
typedef __attribute__((address_space(3))) float lds_float_t;

// Catmull-Rom weights: u0=-.5t+t^2-.5t^3, u1=1-2.5t^2+1.5t^3, u2=.5t+2t^2-1.5t^3, u3=-.5t^2+.5t^3
__device__ __forceinline__ void cr_w(float t, float w[4]) {
    float t2 = t * t, t3 = t2 * t;
    w[0] = fmaf(-0.5f, t3, t2 - 0.5f * t);
    w[1] = fmaf( 1.5f, t3, fmaf(-2.5f, t2, 1.0f));
    w[2] = fmaf(-1.5f, t3, fmaf( 2.0f, t2, 0.5f * t));
    w[3] = fmaf( 0.5f, t3, -0.5f * t2);
}

__global__ __launch_bounds__(BLK) void hashenc_kernel(
    const float* __restrict__ x,       // (npts, 2)
    const float* __restrict__ ht,      // (T, 2, 16): addr = e*32 + f*16 + l
    float* __restrict__ out,           // (npts, 32)
    int npts)
{
    __shared__ __align__(16) float tile[BLK * 32];   // block output tile, layout == global

    const int tid  = threadIdx.x;
    const int gid0 = blockIdx.x * BLK;
    int n  = gid0 + tid;
    int nc = (n < npts) ? n : (npts - 1);   // clamp (keeps barrier convergent)

    const float px = x[2 * nc + 0];
    const float py = x[2 * nc + 1];

    #pragma unroll 1
    for (int l = 0; l < LVLS; ++l) {
        const float nl = c_NL[l];
        const float xs = px * nl, ys = py * nl;
        const int   ix = (int)xs,  iy = (int)ys;        // x,y in [0,1): trunc == floor
        const float tx = xs - (float)ix;
        const float ty = ys - (float)iy;

        float u[4], v[4];
        cr_w(tx, u);
        cr_w(ty, v);

        // h = gx ^ (gy * prime);  idx = h & (T-1)
        const unsigned hx0 = (unsigned)ix;
        const unsigned hy0 = (unsigned)iy * HPRIME;

        float f0[16], f1[16];
        #pragma unroll
        for (int j = 0; j < 4; ++j) {
            const unsigned hy = hy0 + (unsigned)j * HPRIME;
            #pragma unroll
            for (int i = 0; i < 4; ++i) {
                const unsigned idx = ((hx0 + (unsigned)i) ^ hy) & TMASK;
                const float* q = ht + ((size_t)idx << 5) + l;
                f0[j * 4 + i] = q[0];    // feature 0, level l
                f1[j * 4 + i] = q[16];   // feature 1, level l (+64B)
            }
        }

        float a0 = 0.0f, a1 = 0.0f;
        #pragma unroll
        for (int j = 0; j < 4; ++j) {
            #pragma unroll
            for (int i = 0; i < 4; ++i) {
                const float w = u[i] * v[j];
                a0 = fmaf(w, f0[j * 4 + i], a0);
                a1 = fmaf(w, f1[j * 4 + i], a1);
            }
        }

        // stage into LDS, point-major (identical linear layout to global tile)
        *(float2*)&tile[tid * 32 + 2 * l] = make_float2(a0, a1);
    }

    __syncthreads();

    // Drain the 32KB tile with coalesced async b128 stores: 8 rounds x 256 lanes x 16B.
    float* gout = out + (size_t)gid0 * 32;
    #pragma unroll
    for (int k = 0; k < 8; ++k) {
        const int e = (k * BLK + tid) * 4;       // float index within tile
        if (gid0 + (e >> 5) < npts) {
#if __has_builtin(__builtin_amdgcn_global_store_async_from_lds_b128)
            __builtin_amdgcn_global_store_async_from_lds_b128(
                (v4i*)(gout + e), (lds_v4i_t*)(tile + e),
                /*offset=*/0, /*cpol=*/0);
#else
            unsigned long long ga = (unsigned long long)(uintptr_t)(gout + e);
            unsigned la = (unsigned)(size_t)(lds_float_t*)(tile + e);
            asm volatile("global_store_async_from_lds_b128 %0, %1, off"
                         :: "v"(ga), "v"(la)
                         : "memory");
#endif
        }
    }

#if __has_builtin(__builtin_amdgcn_s_wait_asynccnt)
    __builtin_amdgcn_s_wait_asynccnt(0);
#else
    asm volatile("s_wait_asynccnt 0x0" ::: "memory");
#endif
}

extern "C" void kernel_launch(void* const* d_in, const int* in_sizes, int n_in,
                              void* d_out, int out_size, void* d_ws, size_t ws_size,
                              hipStream_t stream) {
    const float* x  = (const float*)d_in[0];   // (N_PTS, 2) float32
    const float* ht = (const float*)d_in[1];   // (T, 2, 16) float32
    float* out = (float*)d_out;                // (N_PTS, 32) float32

    const int npts = in_sizes[0] / 2;
    const int blocks = (npts + BLK - 1) / BLK;
    hipLaunchKernelGGL(hashenc_kernel, dim3(blocks), dim3(BLK), 0, stream,
                       x, ht, out, npts);
}